// ValueDiscriminationLoss_87024627352243
// MI455X (gfx1250) — compile-verified
//
#include <hip/hip_runtime.h>
#include <math.h>

#define MARGIN_F 0.1f
#define S_LEN 2048
#define TILES_J (S_LEN / 16)
#define PAIR_BLOCK 128            // 4 waves of 32
#define WAVES_PER_BLOCK 4
#define BLOCKS_PER_BATCH ((S_LEN / 16) / WAVES_PER_BLOCK)   // 32

typedef __attribute__((ext_vector_type(2))) float v2f;
typedef __attribute__((ext_vector_type(8))) float v8f;

// Workspace layout (floats):
//  [0]=sum_mask [1]=sum_v [2]=sum_r [3]=sum_v2 [4]=sum_r2 [5]=sum_vr
//  [6..13]  = valid[b]     (per-batch mask count)
//  [14..21] = batchloss[b] (per-batch Σ_i mask_i*(rowsum_i - MARGIN))
#define WS_NFLOATS 22

__global__ void vdl_init_ws(float* ws) {
    int t = threadIdx.x;
    if (t < WS_NFLOATS) ws[t] = 0.0f;
}

// Global moment sums over masked values/rewards (for corr, mean, std).
__global__ void vdl_stats(const float* __restrict__ v, const float* __restrict__ r,
                          const int* __restrict__ m, float* __restrict__ ws, int n) {
    int idx = blockIdx.x * blockDim.x + threadIdx.x;
    float mf = 0.f, mv = 0.f, mr = 0.f;
    if (idx < n) {
        mf = (float)m[idx];
        mv = v[idx] * mf;
        mr = r[idx] * mf;
    }
    float s0 = mf, s1 = mv, s2 = mr, s3 = mv * mv, s4 = mr * mr, s5 = mv * mr;
    // wave32 butterfly reduction
    for (int off = 16; off > 0; off >>= 1) {
        s0 += __shfl_xor(s0, off);
        s1 += __shfl_xor(s1, off);
        s2 += __shfl_xor(s2, off);
        s3 += __shfl_xor(s3, off);
        s4 += __shfl_xor(s4, off);
        s5 += __shfl_xor(s5, off);
    }
    if ((threadIdx.x & 31) == 0) {
        atomicAdd(&ws[0], s0);
        atomicAdd(&ws[1], s1);
        atomicAdd(&ws[2], s2);
        atomicAdd(&ws[3], s3);
        atomicAdd(&ws[4], s4);
        atomicAdd(&ws[5], s5);
    }
}

// All-pairs hinge loss. One wave handles one 16-row i-tile, loops over all
// 16-col j-tiles. dv/dr tiles built with V_WMMA_F32_16X16X4_F32:
//   dv = A*B,  A=[v_i, 1, 0, 0] (16x4),  B=[1; -v_j; 0; 0] (4x16)
__global__ void __launch_bounds__(PAIR_BLOCK)
vdl_pairs(const float* __restrict__ v, const float* __restrict__ r,
          const int* __restrict__ m, float* __restrict__ ws) {
    __shared__ float vsh[S_LEN];
    __shared__ float rsh[S_LEN];

    const int b   = blockIdx.x / BLOCKS_PER_BATCH;
    const int tid = threadIdx.x;

    // Stage masked batch row into LDS (values/rewards pre-multiplied by mask,
    // exactly as the reference does before forming pairwise differences).
    for (int t = tid; t < S_LEN; t += PAIR_BLOCK) {
        float mf = (float)m[b * S_LEN + t];
        vsh[t] = v[b * S_LEN + t] * mf;
        rsh[t] = r[b * S_LEN + t] * mf;
    }
    __syncthreads();

    const int wave = tid >> 5;
    const int lane = tid & 31;
    const int half = lane >> 4;   // 0: lanes 0-15 (K=0,1), 1: lanes 16-31 (K=2,3)
    const int l16  = lane & 15;

    const int i_tile = (blockIdx.x % BLOCKS_PER_BATCH) * WAVES_PER_BLOCK + wave;
    const int i0 = i_tile << 4;

    const float one = half ? 0.f : 1.f;
    // A-matrix (16x4, K=0: v_i, K=1: ones, K=2..3: zero)
    v2f a_dv; a_dv.x = half ? 0.f : vsh[i0 + l16]; a_dv.y = one;
    v2f a_dr; a_dr.x = half ? 0.f : rsh[i0 + l16]; a_dr.y = one;

    v8f zero = {};
    v8f acc  = {};

    for (int jt = 0; jt < TILES_J; ++jt) {
        const int j0 = jt << 4;
        // B-matrix (4x16, row0: ones, row1: -v_j, rows 2..3: zero)
        v2f b_dv; b_dv.x = one; b_dv.y = half ? 0.f : -vsh[j0 + l16];
        v2f b_dr; b_dr.x = one; b_dr.y = half ? 0.f : -rsh[j0 + l16];

        v8f dv = __builtin_amdgcn_wmma_f32_16x16x4_f32(
            false, a_dv, false, b_dv, (short)0, zero, false, false);
        v8f dr = __builtin_amdgcn_wmma_f32_16x16x4_f32(
            false, a_dr, false, b_dr, (short)0, zero, false, false);

#pragma unroll
        for (int k = 0; k < 8; ++k) {
            float d = dr[k];
            float s = (d > 0.f) ? 1.f : ((d < 0.f) ? -1.f : 0.f);
            float h = fmaxf(fmaf(dv[k], -s, MARGIN_F), 0.f);
            acc[k] += h;
        }
    }

    // Row sums: reduce over N (16 lanes within each half). D layout: VGPR k
    // holds row M=k in lanes 0-15 and row M=k+8 in lanes 16-31.
    float contrib = 0.f, validp = 0.f;
#pragma unroll
    for (int k = 0; k < 8; ++k) {
        float rs = acc[k];
        rs += __shfl_xor(rs, 1);
        rs += __shfl_xor(rs, 2);
        rs += __shfl_xor(rs, 4);
        rs += __shfl_xor(rs, 8);
        if (l16 == 0) {
            int row = i0 + k + (half << 3);
            float mf = (float)m[b * S_LEN + row];
            contrib += mf * (rs - MARGIN_F);  // remove diagonal term (= MARGIN)
            validp  += mf;
        }
    }
    float c_hi = __shfl(contrib, 16);
    float v_hi = __shfl(validp, 16);
    if (lane == 0) {
        atomicAdd(&ws[14 + b], contrib + c_hi);
        atomicAdd(&ws[6 + b],  validp + v_hi);
    }
}

__global__ void vdl_finalize(const float* __restrict__ ws, float* __restrict__ out,
                             int B, int S) {
    if (threadIdx.x != 0 || blockIdx.x != 0) return;
    const float n = (float)B * (float)S;
    const float sum_mask = ws[0], sum_v = ws[1], sum_r = ws[2];
    const float sum_v2 = ws[3], sum_r2 = ws[4], sum_vr = ws[5];

    float loss = 0.f;
    for (int b = 0; b < B; ++b) {
        float valid = fmaxf(ws[6 + b], 1.f);
        loss += ws[14 + b] / ((float)(S - 1) * valid);
    }
    loss /= (float)B;

    const float cnt = fmaxf(sum_mask, 1.f);
    const float mv = sum_v / cnt, mr = sum_r / cnt;
    const float num = sum_vr - mv * sum_r - mr * sum_v + mv * mr * sum_mask;
    const float dvv = fmaxf(sum_v2 - 2.f * mv * sum_v + mv * mv * sum_mask, 0.f);
    const float drr = fmaxf(sum_r2 - 2.f * mr * sum_r + mr * mr * sum_mask, 0.f);
    const float den = sqrtf(dvv * drr);
    const float corr = (den > 0.f) ? (num / den) : 0.f;

    const float mean_v = sum_v / n;
    const float std_v  = sqrtf(fmaxf((sum_v2 - sum_v * sum_v / n) / (n - 1.f), 0.f));
    const float mean_r = sum_r / n;
    const float std_r  = sqrtf(fmaxf((sum_r2 - sum_r * sum_r / n) / (n - 1.f), 0.f));

    out[0] = loss;
    out[1] = corr;
    out[2] = mean_v;
    out[3] = std_v;
    out[4] = mean_r;
    out[5] = std_r;
}

extern "C" void kernel_launch(void* const* d_in, const int* in_sizes, int n_in,
                              void* d_out, int out_size, void* d_ws, size_t ws_size,
                              hipStream_t stream) {
    const float* values  = (const float*)d_in[0];
    const float* rewards = (const float*)d_in[1];
    const int*   mask    = (const int*)d_in[2];
    float* ws  = (float*)d_ws;
    float* out = (float*)d_out;

    const int n = in_sizes[0];          // B*S
    const int S = S_LEN;
    const int B = n / S;

    vdl_init_ws<<<1, 32, 0, stream>>>(ws);
    vdl_stats<<<(n + 255) / 256, 256, 0, stream>>>(values, rewards, mask, ws, n);
    vdl_pairs<<<B * BLOCKS_PER_BATCH, PAIR_BLOCK, 0, stream>>>(values, rewards, mask, ws);
    vdl_finalize<<<1, 1, 0, stream>>>(ws, out, B, S);
}